// RecurrentGCN_61426622267687
// MI455X (gfx1250) — compile-verified
//
#include <hip/hip_runtime.h>
#include <math.h>

typedef __attribute__((ext_vector_type(2))) float v2f;
typedef __attribute__((ext_vector_type(8))) float v8f;

#define NN 30000
#define NE 480000
#define DF 256
#define DH 256

// ---------------- scatter / gather kernels ----------------

// x_aug[src[e]][d] += x[snf[e]][d]; x_aug[dst[e]][d] += x[dnf[e]][d]
__global__ void aug_scatter(const float* __restrict__ x,
                            const int* __restrict__ src, const int* __restrict__ dst,
                            const int* __restrict__ snf, const int* __restrict__ dnf,
                            float* __restrict__ x_aug) {
    int t = blockIdx.x * blockDim.x + threadIdx.x;
    if (t >= NE * DF) return;           // 122,880,000 < 2^31
    int d = t & (DF - 1);
    int e = t >> 8;
    float a = x[snf[e] * DF + d];
    atomicAdd(&x_aug[src[e] * DF + d], a);
    float b = x[dnf[e] * DF + d];
    atomicAdd(&x_aug[dst[e] * DF + d], b);
}

__global__ void deg_scatter(const int* __restrict__ src, const float* __restrict__ ew,
                            float* __restrict__ deg) {
    int e = blockIdx.x * blockDim.x + threadIdx.x;
    if (e >= NE) return;
    atomicAdd(&deg[src[e]], ew[e]);
}

__global__ void dinv_kernel(const float* __restrict__ deg, float* __restrict__ dinv) {
    int n = blockIdx.x * blockDim.x + threadIdx.x;
    if (n >= NN) return;
    float d = deg[n];
    dinv[n] = (d > 0.0f) ? rsqrtf(fmaxf(d, 1e-12f)) : 0.0f;
}

__global__ void norm_kernel(const int* __restrict__ src, const int* __restrict__ dst,
                            const float* __restrict__ ew, const float* __restrict__ dinv,
                            float* __restrict__ norm) {
    int e = blockIdx.x * blockDim.x + threadIdx.x;
    if (e >= NE) return;
    norm[e] = -dinv[src[e]] * ew[e] * dinv[dst[e]];
}

// hagg[dst[e]][d] += norm[e] * h[src[e]][d]
__global__ void hagg_scatter(const int* __restrict__ src, const int* __restrict__ dst,
                             const float* __restrict__ norm, const float* __restrict__ h,
                             float* __restrict__ hagg) {
    int t = blockIdx.x * blockDim.x + threadIdx.x;
    if (t >= NE * DH) return;
    int d = t & (DH - 1);
    int e = t >> 8;
    atomicAdd(&hagg[dst[e] * DH + d], norm[e] * h[src[e] * DH + d]);
}

// ---------------- weight packing (fragment-swizzled B) ----------------
// Swizzled layout: for each 4x16 B tile (ktile = k/4, ntile = n/16), store the
// 32-lane WMMA B fragment contiguously: lane L holds {B[k0+2*half][n0+lm],
// B[k0+2*half+1][n0+lm]} with half=L>>4, lm=L&15.  Fragment load = 1x b64/lane.
//
// Gates: virtual B is 768x1024 (rows 0..255 = W_z, 256..511 = Wc_z[0],
// 512..767 = Wc_z[1]; column block z*256 holds gate z).
__global__ void pack_Bs_gates(const float* __restrict__ W_i, const float* __restrict__ W_f,
                              const float* __restrict__ W_g, const float* __restrict__ W_o,
                              const float* __restrict__ Wc_i, const float* __restrict__ Wc_f,
                              const float* __restrict__ Wc_g, const float* __restrict__ Wc_o,
                              float* __restrict__ Bs) {
    int t = blockIdx.x * blockDim.x + threadIdx.x;
    if (t >= 768 * 1024) return;
    int j    = t & 1;
    int pair = t >> 1;
    int lane = pair & 31;
    int tile = pair >> 5;
    int ntile = tile & 63;        // N/16 = 64 tiles
    int ktile = tile >> 6;
    int half = lane >> 4, lm = lane & 15;
    int k   = ktile * 4 + 2 * half + j;
    int col = ntile * 16 + lm;
    int z = col >> 8, jn = col & 255;
    const float* W  = (z == 0) ? W_i  : (z == 1) ? W_f  : (z == 2) ? W_g  : W_o;
    const float* Wc = (z == 0) ? Wc_i : (z == 1) ? Wc_f : (z == 2) ? Wc_g : Wc_o;
    float v;
    if (k < 256)      v = W[k * 256 + jn];
    else if (k < 512) v = Wc[(k - 256) * 256 + jn];
    else              v = Wc[65536 + (k - 512) * 256 + jn];
    Bs[t] = v;
}

// Final linear: B is W_lin (256x256 row-major) -> swizzled
__global__ void pack_Bs_lin(const float* __restrict__ W_lin, float* __restrict__ Bs) {
    int t = blockIdx.x * blockDim.x + threadIdx.x;
    if (t >= 256 * 256) return;
    int j    = t & 1;
    int pair = t >> 1;
    int lane = pair & 31;
    int tile = pair >> 5;
    int ntile = tile & 15;        // N/16 = 16 tiles
    int ktile = tile >> 4;
    int half = lane >> 4, lm = lane & 15;
    int k   = ktile * 4 + 2 * half + j;
    int col = ntile * 16 + lm;
    Bs[t] = W_lin[k * 256 + col];
}

// ---------------- fp32 WMMA GEMM ----------------
// C[M,N] = concat_K(A0,A1,A2)[M,K] @ B[K,N] (+ bias[N]).
// A segments: 256 columns each, row-major with row stride 256.
// B is fragment-swizzled (see pack kernels).  One wave computes a 16x128
// strip of C (8 WMMA tiles, A fragment reused 8x).  blockDim = 256 (8 waves).
__global__ void __launch_bounds__(256)
wmma_gemm_f32(const float* __restrict__ A0, const float* __restrict__ A1,
              const float* __restrict__ A2, const float* __restrict__ Bs,
              float* __restrict__ C, const float* __restrict__ bias,
              int M, int N, int K) {
    const int lane = threadIdx.x & 31;
    const int wid  = blockIdx.x * 8 + (threadIdx.x >> 5);
    const int nstrips = N >> 7;                 // 128 columns per wave
    const int totalWaves = (M >> 4) * nstrips;
    if (wid >= totalWaves) return;              // wave-uniform: EXEC stays all-1s
    const int mt = wid / nstrips;
    const int ns = wid - mt * nstrips;
    const int m0 = mt << 4;
    const int n0 = ns << 7;
    const int half = lane >> 4;
    const int lm   = lane & 15;
    const int ntiles = N >> 4;
    const int ntile0 = n0 >> 4;

    v8f acc[8];
#pragma unroll
    for (int i = 0; i < 8; ++i) acc[i] = (v8f){};

    for (int seg = 0; seg * 256 < K; ++seg) {
        const float* A = (seg == 0) ? A0 : (seg == 1) ? A1 : A2;
        const float* arow = A + (m0 + lm) * 256;
        for (int kk = 0; kk < 256; kk += 4) {
            const int k0 = seg * 256 + kk;
            // A fragment (one b64 per lane)
            v2f a = *(const v2f*)(arow + kk + 2 * half);
            // 8 B fragments, each one contiguous b64 per lane
            const v2f* bbase = (const v2f*)Bs + ((size_t)(k0 >> 2) * ntiles + ntile0) * 32 + lane;
#pragma unroll
            for (int i = 0; i < 8; ++i) {
                v2f b = bbase[i * 32];
                acc[i] = __builtin_amdgcn_wmma_f32_16x16x4_f32(
                    false, a, false, b, (short)0, acc[i], false, false);
            }
        }
    }

    // D layout: VGPR v -> row m0 + v + 8*half, col n0 + 16*i + lm
#pragma unroll
    for (int i = 0; i < 8; ++i) {
        const int col = n0 + 16 * i + lm;
        const float bv = bias ? bias[col] : 0.0f;
#pragma unroll
        for (int v = 0; v < 8; ++v) {
            const int row = m0 + v + 8 * half;
            C[row * N + col] = acc[i][v] + bv;
        }
    }
}

// ---------------- elementwise LSTM gates ----------------
__global__ void gates_kernel(const float* __restrict__ G, const float* __restrict__ c,
                             const float* __restrict__ bc_i, const float* __restrict__ bc_f,
                             const float* __restrict__ bc_g, const float* __restrict__ bc_o,
                             const float* __restrict__ b_i, const float* __restrict__ b_f,
                             const float* __restrict__ b_g, const float* __restrict__ b_o,
                             const float* __restrict__ w_ci, const float* __restrict__ w_cf,
                             const float* __restrict__ w_co,
                             float* __restrict__ out_h0, float* __restrict__ out_cnew,
                             float* __restrict__ rbuf) {
    int t = blockIdx.x * blockDim.x + threadIdx.x;
    if (t >= NN * DH) return;
    int j = t & 255;
    int row = t >> 8;
    const float* g = G + row * 1024;
    float cv = c[t];
    float pi = g[j]       + bc_i[j] + b_i[j] + w_ci[j] * cv;
    float pf = g[256 + j] + bc_f[j] + b_f[j] + w_cf[j] * cv;
    float pg = g[512 + j] + bc_g[j] + b_g[j];
    float iv = 1.0f / (1.0f + expf(-pi));
    float fv = 1.0f / (1.0f + expf(-pf));
    float gv = tanhf(pg);
    float cn = fv * cv + iv * gv;
    float po = g[768 + j] + bc_o[j] + b_o[j] + w_co[j] * cn;
    float ov = 1.0f / (1.0f + expf(-po));
    float h0 = ov * tanhf(cn);
    out_h0[t]   = h0;
    out_cnew[t] = cn;
    rbuf[t]     = fmaxf(h0, 0.0f);
}

// ---------------- launch ----------------
extern "C" void kernel_launch(void* const* d_in, const int* in_sizes, int n_in,
                              void* d_out, int out_size, void* d_ws, size_t ws_size,
                              hipStream_t stream) {
    // inputs in setup_inputs() order
    const float* x    = (const float*)d_in[0];
    const int*   ei   = (const int*)  d_in[1];   // (2,E)
    const float* ew   = (const float*)d_in[2];
    const float* h    = (const float*)d_in[3];
    const float* c    = (const float*)d_in[4];
    const int*   snf  = (const int*)  d_in[5];
    const int*   dnf  = (const int*)  d_in[6];
    const float* W_i  = (const float*)d_in[7];
    const float* W_f  = (const float*)d_in[8];
    const float* W_g  = (const float*)d_in[9];
    const float* W_o  = (const float*)d_in[10];
    const float* Wc_i = (const float*)d_in[11];
    const float* Wc_f = (const float*)d_in[12];
    const float* Wc_g = (const float*)d_in[13];
    const float* Wc_o = (const float*)d_in[14];
    const float* bc_i = (const float*)d_in[15];
    const float* bc_f = (const float*)d_in[16];
    const float* bc_g = (const float*)d_in[17];
    const float* bc_o = (const float*)d_in[18];
    const float* b_i  = (const float*)d_in[19];
    const float* b_f  = (const float*)d_in[20];
    const float* b_g  = (const float*)d_in[21];
    const float* b_o  = (const float*)d_in[22];
    const float* b_lin= (const float*)d_in[23];
    const float* w_ci = (const float*)d_in[24];
    const float* w_cf = (const float*)d_in[25];
    const float* w_co = (const float*)d_in[26];
    const float* W_lin= (const float*)d_in[27];

    const int* src = ei;
    const int* dst = ei + NE;

    // workspace layout (floats)
    float* ws    = (float*)d_ws;
    float* x_aug = ws;                       // NN*DF   = 7,680,000
    float* hagg  = x_aug + NN * DF;          // NN*DH   = 7,680,000
    float* deg   = hagg + NN * DH;           // NN
    float* dinv  = deg + NN;                 // NN
    float* norm  = dinv + NN;                // NE
    float* Bs    = norm + NE;                // 768*1024 = 786,432 (swizzled)
    float* BsLin = Bs + 768 * 1024;          // 256*256  = 65,536  (swizzled)
    float* G     = BsLin + 256 * 256;        // NN*1024 = 30,720,000
    float* rbuf  = G + NN * 1024;            // NN*DH   = 7,680,000

    float* out_hout = (float*)d_out;               // h_out
    float* out_h0   = out_hout + NN * DH;          // h0
    float* out_cn   = out_h0 + NN * DH;            // c_new

    // init
    hipMemcpyAsync(x_aug, x, (size_t)NN * DF * sizeof(float), hipMemcpyDeviceToDevice, stream);
    hipMemsetAsync(hagg, 0, (size_t)NN * DH * sizeof(float), stream);
    hipMemsetAsync(deg, 0, (size_t)NN * sizeof(float), stream);

    const int B256 = 256;
    // 1) neighbor-feature augmentation
    aug_scatter<<<(NE * DF + B256 - 1) / B256, B256, 0, stream>>>(x, src, dst, snf, dnf, x_aug);
    // 2) degree / norm
    deg_scatter<<<(NE + B256 - 1) / B256, B256, 0, stream>>>(src, ew, deg);
    dinv_kernel<<<(NN + B256 - 1) / B256, B256, 0, stream>>>(deg, dinv);
    norm_kernel<<<(NE + B256 - 1) / B256, B256, 0, stream>>>(src, dst, ew, dinv, norm);
    // 3) Chebyshev T1 aggregation
    hagg_scatter<<<(NE * DH + B256 - 1) / B256, B256, 0, stream>>>(src, dst, norm, h, hagg);
    // 4) pack weights into fragment-swizzled B layouts
    pack_Bs_gates<<<(768 * 1024) / B256, B256, 0, stream>>>(W_i, W_f, W_g, W_o,
                                                            Wc_i, Wc_f, Wc_g, Wc_o, Bs);
    pack_Bs_lin<<<(256 * 256) / B256, B256, 0, stream>>>(W_lin, BsLin);
    // 5) fused gate GEMM: G = [x_aug | h | hagg] @ B   (30000x768 @ 768x1024)
    {
        int waves = (NN / 16) * (1024 / 128);         // 15000 waves
        wmma_gemm_f32<<<(waves + 7) / 8, 256, 0, stream>>>(x_aug, h, hagg, Bs, G, nullptr,
                                                           NN, 1024, 768);
    }
    // 6) elementwise LSTM gates + peepholes
    gates_kernel<<<(NN * DH + B256 - 1) / B256, B256, 0, stream>>>(
        G, c, bc_i, bc_f, bc_g, bc_o, b_i, b_f, b_g, b_o, w_ci, w_cf, w_co,
        out_h0, out_cn, rbuf);
    // 7) h_out = relu(h0) @ W_lin + b_lin   (30000x256 @ 256x256)
    {
        int waves = (NN / 16) * (256 / 128);          // 3750 waves
        wmma_gemm_f32<<<(waves + 7) / 8, 256, 0, stream>>>(rbuf, rbuf, rbuf, BsLin, out_hout,
                                                           b_lin, NN, 256, 256);
    }
}